// SGCEncoder_80985903333764
// MI455X (gfx1250) — compile-verified
//
#include <hip/hip_runtime.h>
#include <hip/hip_bf16.h>

typedef float v2f __attribute__((ext_vector_type(2)));
typedef float v8f __attribute__((ext_vector_type(8)));

// ---------------------------------------------------------------------------
// Degree / normalization
// ---------------------------------------------------------------------------
__global__ void sgc_deg_init(float* __restrict__ deg, int n) {
    int i = blockIdx.x * blockDim.x + threadIdx.x;
    if (i < n) deg[i] = 1.0f;            // self-loop contributes 1 to every node
}

__global__ void sgc_deg_count(const int* __restrict__ col, float* __restrict__ deg, int ne) {
    int e = blockIdx.x * blockDim.x + threadIdx.x;
    if (e < ne) atomicAdd(&deg[col[e]], 1.0f);
}

__global__ void sgc_dinv(float* __restrict__ deg, int n) {
    int i = blockIdx.x * blockDim.x + threadIdx.x;
    if (i < n) deg[i] = rsqrtf(deg[i]);  // deg >= 1 always (self-loops)
}

// ---------------------------------------------------------------------------
// Self-loop term: hout[i][:] = dinv[i]^2 * hin[i][:]
// One thread per 4 channels (float4), 32 threads per node.
// ---------------------------------------------------------------------------
__global__ void sgc_selfloop(const float* __restrict__ dinv,
                             const float* __restrict__ hin,
                             float* __restrict__ hout, int n) {
    int gid  = blockIdx.x * blockDim.x + threadIdx.x;
    int node = gid >> 5;
    if (node >= n) return;
    int c4 = gid & 31;
    float d = dinv[node];
    float w = d * d;
    float4 v = ((const float4*)(hin + (size_t)node * 128))[c4];
    float4 o;
    o.x = w * v.x; o.y = w * v.y; o.z = w * v.z; o.w = w * v.w;
    ((float4*)(hout + (size_t)node * 128))[c4] = o;
}

// ---------------------------------------------------------------------------
// Edge scatter: one wave32 per edge; lane handles 4 channels.
// hout[i][:] += dinv[j]*dinv[i] * hin[j][:]
// ---------------------------------------------------------------------------
__global__ void sgc_edge_scatter(const int* __restrict__ row,
                                 const int* __restrict__ col,
                                 const float* __restrict__ dinv,
                                 const float* __restrict__ hin,
                                 float* __restrict__ hout, int ne) {
    int gid = blockIdx.x * blockDim.x + threadIdx.x;
    int e   = gid >> 5;                  // wave-uniform
    if (e >= ne) return;
    int lane = gid & 31;
    int j = row[e];                      // source
    int i = col[e];                      // target
    float w = dinv[j] * dinv[i];
    float4 v = ((const float4*)(hin + (size_t)j * 128))[lane];   // global_load_b128
    float* dst = hout + (size_t)i * 128 + (size_t)lane * 4;
    atomicAdd(dst + 0, w * v.x);
    atomicAdd(dst + 1, w * v.y);
    atomicAdd(dst + 2, w * v.z);
    atomicAdd(dst + 3, w * v.w);
}

// ---------------------------------------------------------------------------
// Final projection with fp32 WMMA: out[N,128] = h[N,128] @ W^T + b
// W is [128,128] row-major (W[n][k]), so B(KxN) column n is W row n.
// Block = 256 threads = 8 waves; block owns one 16-row M tile, wave w owns
// N-tile w (16 columns). K loop: 32 x V_WMMA_F32_16X16X4_F32.
// ---------------------------------------------------------------------------
__global__ void sgc_gemm_wmma(const float* __restrict__ h,
                              const float* __restrict__ W,
                              const float* __restrict__ bias,
                              float* __restrict__ out, int n_nodes) {
    const int wave = threadIdx.x >> 5;
    const int lane = threadIdx.x & 31;
    const int m0   = blockIdx.x * 16;
    if (m0 >= n_nodes) return;           // block-uniform: EXEC stays all-ones
    const int n0 = wave * 16;
    const int lo = lane & 15;            // M (for A) / N (for B,C,D)
    const int hi = lane >> 4;            // K half-select for A/B, M+8 for C/D

    // A: lane holds row m0+lo, K elements {k+2*hi, k+2*hi+1}
    const float* arow = h + (size_t)(m0 + lo) * 128 + 2 * hi;
    // B: lane holds column n0+lo of W^T == row n0+lo of W, same K pattern
    const float* wrow = W + (size_t)(n0 + lo) * 128 + 2 * hi;

    v8f c = {0.f, 0.f, 0.f, 0.f, 0.f, 0.f, 0.f, 0.f};
#pragma unroll
    for (int k = 0; k < 128; k += 4) {
        v2f a, b;
        a.x = arow[k];  a.y = arow[k + 1];
        b.x = wrow[k];  b.y = wrow[k + 1];
        // (neg_a, A, neg_b, B, c_mod, C, reuse_a, reuse_b)
        c = __builtin_amdgcn_wmma_f32_16x16x4_f32(
                false, a, false, b, (short)0, c, false, false);
    }

    // C/D layout: VGPR v -> M = v + 8*hi, N = lo
    const float bb = bias[n0 + lo];
    float* orow = out + ((size_t)(m0 + hi * 8)) * 128 + n0 + lo;
#pragma unroll
    for (int v = 0; v < 8; ++v) {
        orow[(size_t)v * 128] = c[v] + bb;
    }
}

// ---------------------------------------------------------------------------
// Launcher
// ---------------------------------------------------------------------------
extern "C" void kernel_launch(void* const* d_in, const int* in_sizes, int n_in,
                              void* d_out, int out_size, void* d_ws, size_t ws_size,
                              hipStream_t stream) {
    const float* x    = (const float*)d_in[0];     // [N,128]
    const int*   ei   = (const int*)d_in[1];       // [2,E]
    const float* W    = (const float*)d_in[2];     // [128,128]
    const float* bias = (const float*)d_in[3];     // [128]

    const int n  = in_sizes[0] / 128;
    const int ne = in_sizes[1] / 2;
    const int* row = ei;        // sources j
    const int* col = ei + ne;   // targets i

    float* ws   = (float*)d_ws;
    size_t npad = ((size_t)n + 255) & ~(size_t)255;
    float* dinv = ws;                          // N floats (deg -> dinv in place)
    float* h1   = ws + npad;                   // N*128 floats
    float* h2   = h1 + (size_t)n * 128;        // N*128 floats
    float* outp = (float*)d_out;

    const int B = 256;
    // normalization
    sgc_deg_init <<<(n  + B - 1) / B, B, 0, stream>>>(dinv, n);
    sgc_deg_count<<<(ne + B - 1) / B, B, 0, stream>>>(col, dinv, ne);
    sgc_dinv     <<<(n  + B - 1) / B, B, 0, stream>>>(dinv, n);

    const long long nthr_node = (long long)n  * 32;
    const long long nthr_edge = (long long)ne * 32;
    // hop 1: x -> h1
    sgc_selfloop    <<<(nthr_node + B - 1) / B, B, 0, stream>>>(dinv, x, h1, n);
    sgc_edge_scatter<<<(nthr_edge + B - 1) / B, B, 0, stream>>>(row, col, dinv, x, h1, ne);
    // hop 2: h1 -> h2
    sgc_selfloop    <<<(nthr_node + B - 1) / B, B, 0, stream>>>(dinv, h1, h2, n);
    sgc_edge_scatter<<<(nthr_edge + B - 1) / B, B, 0, stream>>>(row, col, dinv, h1, h2, ne);
    // projection: h2 @ W^T + b -> out
    sgc_gemm_wmma<<<(n + 15) / 16, B, 0, stream>>>(h2, W, bias, outp, n);
}